// HGCN_27745488732639
// MI455X (gfx1250) — compile-verified
//
#include <hip/hip_runtime.h>

#define N_NODES 100000
#define N_EDGES 1600000
#define DIM_IN  128
#define DIM_HID 64
#define DIM_OUT 40
#define MAXN 0.999f      /* (1-1e-3)/sqrt(c), c=1 */
#define EPSF 1e-15f

typedef float v2f __attribute__((ext_vector_type(2)));
typedef float v8f __attribute__((ext_vector_type(8)));

__device__ __forceinline__ float red16_add(float v) {
  v += __shfl_xor(v, 1, 32);
  v += __shfl_xor(v, 2, 32);
  v += __shfl_xor(v, 4, 32);
  v += __shfl_xor(v, 8, 32);
  return v;  // sum within each 16-lane half
}
__device__ __forceinline__ float red16_max(float v) {
  v = fmaxf(v, __shfl_xor(v, 1, 32));
  v = fmaxf(v, __shfl_xor(v, 2, 32));
  v = fmaxf(v, __shfl_xor(v, 4, 32));
  v = fmaxf(v, __shfl_xor(v, 8, 32));
  return v;
}

// ---------------------------------------------------------------------------
// Encoder: xh = proj(expmap0(x)); h = proj(mobius_matvec(W1,xh));
//          h = proj(mobius_add(h, proj(expmap0(b1)))); ht = logmap0(h)
// One wave per 16-row tile. GEMM on raw x via V_WMMA_F32_16X16X4_F32;
// all hyperbolic maps collapse to per-row scalars applied in the epilogue.
// ---------------------------------------------------------------------------
__global__ __launch_bounds__(256) void hgcn_enc_kernel(
    const float* __restrict__ x, const float* __restrict__ W1,
    const float* __restrict__ b1, float* __restrict__ ht) {
  const int wave = (blockIdx.x * blockDim.x + threadIdx.x) >> 5;
  if (wave >= N_NODES / 16) return;             // wave-uniform exit
  const int lane = threadIdx.x & 31;
  const int l16  = lane & 15;
  const int half = lane >> 4;
  const int row0 = wave * 16;

  // bias_h = proj(expmap0(b1)): per-element scale of b1 (uniform scalar loads)
  float nb2 = 0.f;
  #pragma unroll 8
  for (int j = 0; j < DIM_HID; ++j) { float bv = b1[j]; nb2 += bv * bv; }
  const float nb     = sqrtf(fmaxf(nb2, 1e-30f));
  const float rb     = fminf(tanhf(nb), MAXN);
  const float bscale = rb / fmaxf(nb, EPSF);
  const float y2c    = rb * rb;                 // |bias_h|^2
  float biash[4];
  #pragma unroll
  for (int nt = 0; nt < 4; ++nt) biash[nt] = bscale * b1[nt * 16 + l16];

  // GEMM: C = x_tile[16x128] @ W1^T[128x64], K-steps of 4 (fp32 WMMA)
  v8f c0 = {}, c1 = {}, c2 = {}, c3 = {};
  float nx2p = 0.f;                             // partial |x_row|^2
  const float* xrow = x + (long long)(row0 + l16) * DIM_IN + 2 * half;
  #pragma unroll 4
  for (int kk = 0; kk < DIM_IN; kk += 4) {
    v2f a = *(const v2f*)(xrow + kk);           // A: K = kk+2*half, +1
    nx2p += a.x * a.x + a.y * a.y;
    const float* wk = W1 + (long long)l16 * DIM_IN + kk + 2 * half;
    v2f b0v = *(const v2f*)(wk + 0 * 16 * DIM_IN);  // B[k][n] = W1[n][k]
    v2f b1v = *(const v2f*)(wk + 1 * 16 * DIM_IN);
    v2f b2v = *(const v2f*)(wk + 2 * 16 * DIM_IN);
    v2f b3v = *(const v2f*)(wk + 3 * 16 * DIM_IN);
    c0 = __builtin_amdgcn_wmma_f32_16x16x4_f32(false, a, false, b0v, (short)0, c0, false, false);
    c1 = __builtin_amdgcn_wmma_f32_16x16x4_f32(false, a, false, b1v, (short)0, c1, false, false);
    c2 = __builtin_amdgcn_wmma_f32_16x16x4_f32(false, a, false, b2v, (short)0, c2, false, false);
    c3 = __builtin_amdgcn_wmma_f32_16x16x4_f32(false, a, false, b3v, (short)0, c3, false, false);
  }
  // full |x_row(l16)|^2 in every lane
  const float nx2 = nx2p + __shfl_xor(nx2p, 16, 32);

  // per-row reductions over C fragments (row r = i + 8*half for this lane)
  float msq[8], dotb[8];
  #pragma unroll
  for (int i = 0; i < 8; ++i) {
    float e0 = c0[i], e1 = c1[i], e2 = c2[i], e3 = c3[i];
    msq[i]  = red16_add(e0 * e0 + e1 * e1 + e2 * e2 + e3 * e3);
    dotb[i] = red16_add(e0 * biash[0] + e1 * biash[1] + e2 * biash[2] + e3 * biash[3]);
  }

  #pragma unroll
  for (int i = 0; i < 8; ++i) {
    const int r = i + (half << 3);
    float nxr  = __shfl(nx2, r, 32);            // |x_row|^2 for row r
    float n    = sqrtf(fmaxf(nxr, 1e-30f));
    float xn   = fminf(tanhf(n), MAXN);         // |xh| after expmap0+proj
    float nmx  = sqrtf(fmaxf(msq[i], 1e-30f));  // |x @ W1^T| (raw)
    float beta = (nmx / fmaxf(n, EPSF)) * atanhf(xn);
    float hn   = fminf(tanhf(beta), MAXN);      // |h| after matvec+proj
    float dl   = hn / fmaxf(nmx, EPSF);         // h = dl * mxraw
    float x2   = hn * hn;
    float xy   = dl * dotb[i];
    float cA   = (1.f + 2.f * xy + y2c) * dl;
    float cB   = 1.f - x2;
    float inv  = 1.f / fmaxf(1.f + 2.f * xy + x2 * y2c, EPSF);
    float h0 = (cA * c0[i] + cB * biash[0]) * inv;
    float h1 = (cA * c1[i] + cB * biash[1]) * inv;
    float h2 = (cA * c2[i] + cB * biash[2]) * inv;
    float h3 = (cA * c3[i] + cB * biash[3]) * inv;
    float n2sq = red16_add(h0 * h0 + h1 * h1 + h2 * h2 + h3 * h3);
    float n2   = sqrtf(fmaxf(n2sq, 1e-30f));
    float nn   = fminf(n2, MAXN);               // proj, then logmap0
    float tau  = atanhf(nn) / fmaxf(n2, EPSF);
    float* o = ht + (long long)(row0 + r) * DIM_HID + l16;
    o[0]  = tau * h0;
    o[16] = tau * h1;
    o[32] = tau * h2;
    o[48] = tau * h3;
  }
}

// ---------------------------------------------------------------------------
__global__ void hgcn_zero_kernel(float* __restrict__ agg, float* __restrict__ deg) {
  int t = blockIdx.x * blockDim.x + threadIdx.x;
  if (t < N_NODES * DIM_HID) agg[t] = 0.f;
  if (t < N_NODES) deg[t] = 0.f;
}

__global__ void hgcn_deg_kernel(const int* __restrict__ ei, float* __restrict__ deg) {
  int e = blockIdx.x * blockDim.x + threadIdx.x;
  if (e >= N_EDGES) return;
  int r = ei[e];
  if ((unsigned)r < (unsigned)N_NODES) atomicAdd(&deg[r], 1.0f);
}

// one 32-lane wave per edge, float2 per lane; agg/ht stay L2-resident
__global__ void hgcn_agg_kernel(const int* __restrict__ ei,
                                const float* __restrict__ ht,
                                const float* __restrict__ deg,
                                float* __restrict__ agg) {
  long long tid = (long long)blockIdx.x * blockDim.x + threadIdx.x;
  int e    = (int)(tid >> 5);
  int lane = (int)(tid & 31);
  if (e >= N_EDGES) return;
  int r = ei[e];
  int c = ei[N_EDGES + e];
  if ((unsigned)r >= (unsigned)N_NODES || (unsigned)c >= (unsigned)N_NODES) return;
  float w = rsqrtf(fmaxf(deg[r], 1.f) * fmaxf(deg[c], 1.f));
  v2f v = *(const v2f*)(ht + (long long)c * DIM_HID + 2 * lane);
  float* dst = agg + (long long)r * DIM_HID + 2 * lane;
  atomicAdd(dst + 0, v.x * w);
  atomicAdd(dst + 1, v.y * w);
}

// ---------------------------------------------------------------------------
// Decoder: h=proj(expmap0(agg)); h=proj(expmap0(relu(logmap0(h))));
//          z=logmap0(h); out = z@Wd^T + bd; log_softmax.
// z = zscale * relu(t_over_n * agg) elementwise (both scalars per row).
// WMMA with N padded 40 -> 48 (zero B frag), masked softmax/stores.
// ---------------------------------------------------------------------------
__global__ __launch_bounds__(256) void hgcn_dec_kernel(
    const float* __restrict__ agg, const float* __restrict__ Wd,
    const float* __restrict__ bd, float* __restrict__ out) {
  const int wave = (blockIdx.x * blockDim.x + threadIdx.x) >> 5;
  if (wave >= N_NODES / 16) return;
  const int lane = threadIdx.x & 31;
  const int l16  = lane & 15;
  const int half = lane >> 4;
  const int row0 = wave * 16;
  const float* arow = agg + (long long)(row0 + l16) * DIM_HID + 2 * half;

  // pass 1: |agg_row|^2  ->  t_over_n = artanh(min(tanh n, maxn)) / n
  float p = 0.f;
  #pragma unroll
  for (int kk = 0; kk < DIM_HID; kk += 4) {
    v2f a = *(const v2f*)(arow + kk);
    p += a.x * a.x + a.y * a.y;
  }
  float n2a = p + __shfl_xor(p, 16, 32);
  float na  = sqrtf(fmaxf(n2a, 1e-30f));
  float t_over_n = atanhf(fminf(tanhf(na), MAXN)) / fmaxf(na, EPSF);  // >= 0

  // pass 2: |relu(t_over_n * agg)|^2 -> zscale
  p = 0.f;
  #pragma unroll
  for (int kk = 0; kk < DIM_HID; kk += 4) {
    v2f a = *(const v2f*)(arow + kk);
    float u0 = fmaxf(t_over_n * a.x, 0.f), u1 = fmaxf(t_over_n * a.y, 0.f);
    p += u0 * u0 + u1 * u1;
  }
  float n2u = p + __shfl_xor(p, 16, 32);
  float nu  = sqrtf(fmaxf(n2u, 1e-30f));
  float zscale = atanhf(fminf(tanhf(nu), MAXN)) / fmaxf(nu, EPSF);
  float ascale = zscale * t_over_n;             // z = ascale * max(agg, 0)

  const int j0 = l16, j1 = 16 + l16, j2 = 32 + l16;
  const bool valid2 = (l16 < DIM_OUT - 32);     // l16 < 8
  v8f c0 = {}, c1 = {}, c2 = {};
  #pragma unroll
  for (int kk = 0; kk < DIM_HID; kk += 4) {
    v2f a = *(const v2f*)(arow + kk);
    v2f af; af.x = ascale * fmaxf(a.x, 0.f); af.y = ascale * fmaxf(a.y, 0.f);
    const float* wk = Wd + kk + 2 * half;       // B[k][j] = Wd[j][k]
    v2f b0v = *(const v2f*)(wk + (long long)j0 * DIM_HID);
    v2f b1v = *(const v2f*)(wk + (long long)j1 * DIM_HID);
    v2f b2v; b2v.x = 0.f; b2v.y = 0.f;
    if (valid2) b2v = *(const v2f*)(wk + (long long)j2 * DIM_HID);
    c0 = __builtin_amdgcn_wmma_f32_16x16x4_f32(false, af, false, b0v, (short)0, c0, false, false);
    c1 = __builtin_amdgcn_wmma_f32_16x16x4_f32(false, af, false, b1v, (short)0, c1, false, false);
    c2 = __builtin_amdgcn_wmma_f32_16x16x4_f32(false, af, false, b2v, (short)0, c2, false, false);
  }

  const float bd0 = bd[j0];
  const float bd1 = bd[j1];
  const float bd2 = valid2 ? bd[j2] : 0.f;
  const float NEG_BIG = -3.0e38f;
  #pragma unroll
  for (int i = 0; i < 8; ++i) {
    const int r = i + (half << 3);
    float v0 = c0[i] + bd0;
    float v1 = c1[i] + bd1;
    float v2 = c2[i] + bd2;
    float m  = red16_max(fmaxf(fmaxf(v0, v1), valid2 ? v2 : NEG_BIG));
    float s  = red16_add(expf(v0 - m) + expf(v1 - m) + (valid2 ? expf(v2 - m) : 0.f));
    float lse = m + logf(s);
    float* o = out + (long long)(row0 + r) * DIM_OUT;
    o[j0] = v0 - lse;
    o[j1] = v1 - lse;
    if (valid2) o[j2] = v2 - lse;
  }
}

// ---------------------------------------------------------------------------
extern "C" void kernel_launch(void* const* d_in, const int* in_sizes, int n_in,
                              void* d_out, int out_size, void* d_ws, size_t ws_size,
                              hipStream_t stream) {
  const float* x  = (const float*)d_in[0];
  const int*   ei = (const int*)d_in[1];
  const float* W1 = (const float*)d_in[2];
  const float* b1 = (const float*)d_in[3];
  const float* Wd = (const float*)d_in[4];
  const float* bd = (const float*)d_in[5];
  float* out = (float*)d_out;

  float* ht  = (float*)d_ws;                       // [N, 64]
  float* agg = ht + (size_t)N_NODES * DIM_HID;     // [N, 64]
  float* deg = agg + (size_t)N_NODES * DIM_HID;    // [N]

  const int tiles = N_NODES / 16;                  // 6250 exactly
  dim3 blk(256);

  hgcn_zero_kernel<<<(N_NODES * DIM_HID + 255) / 256, blk, 0, stream>>>(agg, deg);
  hgcn_enc_kernel<<<(tiles * 32 + 255) / 256, blk, 0, stream>>>(x, W1, b1, ht);
  hgcn_deg_kernel<<<(N_EDGES + 255) / 256, blk, 0, stream>>>(ei, deg);
  hgcn_agg_kernel<<<(int)(((long long)N_EDGES * 32 + 255) / 256), blk, 0, stream>>>(ei, ht, deg, agg);
  hgcn_dec_kernel<<<(tiles * 32 + 255) / 256, blk, 0, stream>>>(agg, Wd, bd, out);
}